// Intention_46488726012118
// MI455X (gfx1250) — compile-verified
//
#include <hip/hip_runtime.h>
#include <hip/hip_bf16.h>

typedef __attribute__((ext_vector_type(2))) float v2f;
typedef __attribute__((ext_vector_type(8))) float v8f;

#define DIM   768
#define NSEQ  4096
#define BATCH 4
#define HEADS 12
#define HDIM  64
#define MTOT  (BATCH * NSEQ)   // 16384

// ---------------------------------------------------------------------------
// Fused QKV projection: {Q,K,V}[M,N] = X[M,K] @ W{q,k,v}[N,K]^T
// Per wave: 16(M) x 32(N) of all three outputs = 6 accumulators.
// Software-pipelined: loads for step k+4 issue before WMMAs of step k, so
// each WMMA group waits on loads issued a full iteration earlier.
// ---------------------------------------------------------------------------
__global__ __launch_bounds__(256) void qkv_gemm(const float* __restrict__ X,
                                                const float* __restrict__ Wq,
                                                const float* __restrict__ Wk,
                                                const float* __restrict__ Wv,
                                                float* __restrict__ Qo,
                                                float* __restrict__ Ko,
                                                float* __restrict__ Vo) {
  const int lane = threadIdx.x & 31;
  const int wave = threadIdx.x >> 5;
  const int half = lane >> 4;      // K pair select: {0,1} vs {2,3}
  const int lid  = lane & 15;
  const int m0 = blockIdx.x * 128 + wave * 16;
  const int n0 = blockIdx.y * 32;

  const float* xrow = X + (size_t)(m0 + lid) * DIM + half * 2;
  const float* bp[6];
  {
    const size_t r0 = (size_t)(n0 + lid) * DIM + half * 2;
    const size_t r1 = (size_t)(n0 + 16 + lid) * DIM + half * 2;
    bp[0] = Wq + r0; bp[1] = Wq + r1;
    bp[2] = Wk + r0; bp[3] = Wk + r1;
    bp[4] = Wv + r0; bp[5] = Wv + r1;
  }

  v8f acc[6] = {};
  v2f a_cur = *(const v2f*)(xrow);
  v2f b_cur[6];
#pragma unroll
  for (int j = 0; j < 6; ++j) b_cur[j] = *(const v2f*)(bp[j]);

  for (int k = 0; k < DIM - 4; k += 4) {
    // prefetch next K-step (7 loads in flight over the WMMA group below)
    const v2f a_nxt = *(const v2f*)(xrow + k + 4);
    v2f b_nxt[6];
#pragma unroll
    for (int j = 0; j < 6; ++j) b_nxt[j] = *(const v2f*)(bp[j] + k + 4);

#pragma unroll
    for (int j = 0; j < 6; ++j)
      acc[j] = __builtin_amdgcn_wmma_f32_16x16x4_f32(false, a_cur, false, b_cur[j],
                                                     (short)0, acc[j], false, false);
    a_cur = a_nxt;
#pragma unroll
    for (int j = 0; j < 6; ++j) b_cur[j] = b_nxt[j];
  }
#pragma unroll
  for (int j = 0; j < 6; ++j)
    acc[j] = __builtin_amdgcn_wmma_f32_16x16x4_f32(false, a_cur, false, b_cur[j],
                                                   (short)0, acc[j], false, false);

#pragma unroll
  for (int jj = 0; jj < 2; ++jj) {
    const int n = n0 + jj * 16 + lid;
#pragma unroll
    for (int r = 0; r < 8; ++r) {
      const size_t o = (size_t)(m0 + r + half * 8) * DIM + n;
      Qo[o] = acc[jj    ][r];
      Ko[o] = acc[jj + 2][r];
      Vo[o] = acc[jj + 4][r];
    }
  }
}

// ---------------------------------------------------------------------------
// Output projection: Y[M,N] = X[M,K] @ W[N,K]^T + bias[N]
// Per wave: 16(M) x 64(N); same one-iteration software pipeline.
// ---------------------------------------------------------------------------
__global__ __launch_bounds__(256) void out_gemm(const float* __restrict__ X,
                                                const float* __restrict__ W,
                                                const float* __restrict__ bias,
                                                float* __restrict__ Y) {
  const int lane = threadIdx.x & 31;
  const int wave = threadIdx.x >> 5;
  const int half = lane >> 4, lid = lane & 15;
  const int m0 = blockIdx.x * 128 + wave * 16;
  const int n0 = blockIdx.y * 64;

  const float* xrow = X + (size_t)(m0 + lid) * DIM + half * 2;
  const float* wr[4];
#pragma unroll
  for (int j = 0; j < 4; ++j)
    wr[j] = W + (size_t)(n0 + j * 16 + lid) * DIM + half * 2;

  v8f acc[4] = {};
  v2f a_cur = *(const v2f*)(xrow);
  v2f b_cur[4];
#pragma unroll
  for (int j = 0; j < 4; ++j) b_cur[j] = *(const v2f*)(wr[j]);

  for (int k = 0; k < DIM - 4; k += 4) {
    const v2f a_nxt = *(const v2f*)(xrow + k + 4);
    v2f b_nxt[4];
#pragma unroll
    for (int j = 0; j < 4; ++j) b_nxt[j] = *(const v2f*)(wr[j] + k + 4);

#pragma unroll
    for (int j = 0; j < 4; ++j)
      acc[j] = __builtin_amdgcn_wmma_f32_16x16x4_f32(false, a_cur, false, b_cur[j],
                                                     (short)0, acc[j], false, false);
    a_cur = a_nxt;
#pragma unroll
    for (int j = 0; j < 4; ++j) b_cur[j] = b_nxt[j];
  }
#pragma unroll
  for (int j = 0; j < 4; ++j)
    acc[j] = __builtin_amdgcn_wmma_f32_16x16x4_f32(false, a_cur, false, b_cur[j],
                                                   (short)0, acc[j], false, false);

#pragma unroll
  for (int j = 0; j < 4; ++j) {
    const int n = n0 + j * 16 + lid;
    const float bv = bias[n];
#pragma unroll
    for (int r = 0; r < 8; ++r)
      Y[(size_t)(m0 + r + half * 8) * DIM + n] = acc[j][r] + bv;
  }
}

// ---------------------------------------------------------------------------
// Per-(b,h) Gram matrices: KK[d,e] = sum_n K[n,d]K[n,e]; KTV[d,e]=sum K[n,d]V[n,e]
// One 512-thread block per (b,h); 16 waves each own one 16x16 tile of both.
// ---------------------------------------------------------------------------
__global__ __launch_bounds__(512) void gram_kernel(const float* __restrict__ Km,
                                                   const float* __restrict__ Vm,
                                                   float* __restrict__ KK,
                                                   float* __restrict__ KTV) {
  const int bh = blockIdx.x;                 // 0..47
  const int b  = bh / HEADS, h = bh % HEADS;
  const int lane = threadIdx.x & 31;
  const int wave = threadIdx.x >> 5;
  const int half = lane >> 4, lid = lane & 15;
  const int Dbase = (wave >> 2) * 16;
  const int Ebase = (wave & 3) * 16;

  const float* kbase = Km + ((size_t)b * NSEQ) * DIM + h * HDIM;
  const float* vbase = Vm + ((size_t)b * NSEQ) * DIM + h * HDIM;

  v8f acck = {}, accv = {};
#pragma unroll 2
  for (int n = 0; n < NSEQ; n += 4) {
    const int nn = n + half * 2;
    const size_t r0 = (size_t)nn * DIM;
    const size_t r1 = (size_t)(nn + 1) * DIM;
    v2f a;  a.x  = kbase[r0 + Dbase + lid]; a.y  = kbase[r1 + Dbase + lid];
    v2f bk; bk.x = kbase[r0 + Ebase + lid]; bk.y = kbase[r1 + Ebase + lid];
    v2f bv; bv.x = vbase[r0 + Ebase + lid]; bv.y = vbase[r1 + Ebase + lid];
    acck = __builtin_amdgcn_wmma_f32_16x16x4_f32(false, a, false, bk,
                                                 (short)0, acck, false, false);
    accv = __builtin_amdgcn_wmma_f32_16x16x4_f32(false, a, false, bv,
                                                 (short)0, accv, false, false);
  }

  float* kko  = KK  + (size_t)bh * HDIM * HDIM;
  float* ktvo = KTV + (size_t)bh * HDIM * HDIM;
#pragma unroll
  for (int r = 0; r < 8; ++r) {
    const int d = Dbase + r + half * 8;
    const int e = Ebase + lid;
    kko [d * HDIM + e] = acck[r];
    ktvo[d * HDIM + e] = accv[r];
  }
}

// ---------------------------------------------------------------------------
// Per-(b,h): inv(KK + alpha*I) @ KTV, then softmax over rows (axis=-2).
// 64 threads: thread t owns column t. Gauss-Jordan without pivoting (SPD).
// ---------------------------------------------------------------------------
__global__ __launch_bounds__(64) void inv_soft_kernel(const float* __restrict__ KK,
                                                      const float* __restrict__ KTV,
                                                      const float* __restrict__ alpha,
                                                      float* __restrict__ ATTN) {
  __shared__ float A[HDIM * HDIM];
  __shared__ float Iv[HDIM * HDIM];
  __shared__ float col[HDIM];

  const int bh = blockIdx.x;
  const int t  = threadIdx.x;
  const float* kk  = KK  + (size_t)bh * HDIM * HDIM;
  const float* ktv = KTV + (size_t)bh * HDIM * HDIM;
  const float al = alpha[0];

  for (int i = 0; i < HDIM; ++i) {
    A [i * HDIM + t] = kk[i * HDIM + t] + ((i == t) ? al : 0.0f);
    Iv[i * HDIM + t] = (i == t) ? 1.0f : 0.0f;
  }
  __syncthreads();

  for (int p = 0; p < HDIM; ++p) {
    col[t] = A[t * HDIM + p];
    __syncthreads();
    const float invp = 1.0f / col[p];
    const float arow = A [p * HDIM + t] * invp;
    const float irow = Iv[p * HDIM + t] * invp;
    A [p * HDIM + t] = arow;
    Iv[p * HDIM + t] = irow;
    for (int r = 0; r < HDIM; ++r) {
      if (r == p) continue;
      const float f = col[r];
      A [r * HDIM + t] -= f * arow;
      Iv[r * HDIM + t] -= f * irow;
    }
    __syncthreads();
  }

  float kcol[HDIM];
#pragma unroll 8
  for (int j = 0; j < HDIM; ++j) kcol[j] = ktv[j * HDIM + t];
  for (int i = 0; i < HDIM; ++i) {
    float s = 0.0f;
#pragma unroll 8
    for (int j = 0; j < HDIM; ++j) s += Iv[i * HDIM + j] * kcol[j];
    A[i * HDIM + t] = s;
  }

  float mx = -3.402823466e38f;
  for (int i = 0; i < HDIM; ++i) mx = fmaxf(mx, A[i * HDIM + t]);
  float sum = 0.0f;
  for (int i = 0; i < HDIM; ++i) {
    const float e = expf(A[i * HDIM + t] - mx);
    A[i * HDIM + t] = e;
    sum += e;
  }
  const float rs = 1.0f / sum;
  float* out = ATTN + (size_t)bh * HDIM * HDIM;
  for (int i = 0; i < HDIM; ++i) out[i * HDIM + t] = A[i * HDIM + t] * rs;
}

// ---------------------------------------------------------------------------
// CTX[b,n,h*64+d] = sum_e Q[b,n,h*64+e] * ATTN[bh][e,d]
// ---------------------------------------------------------------------------
__global__ __launch_bounds__(128) void apply_attn(const float* __restrict__ Q,
                                                  const float* __restrict__ ATTN,
                                                  float* __restrict__ CTX) {
  const int bh = blockIdx.y;
  const int b  = bh / HEADS, h = bh % HEADS;
  const int lane = threadIdx.x & 31;
  const int wave = threadIdx.x >> 5;
  const int half = lane >> 4, lid = lane & 15;
  const int m0 = blockIdx.x * 16;
  const int n0 = wave * 16;

  const float* qrow = Q + ((size_t)b * NSEQ + m0 + lid) * DIM + h * HDIM;
  const float* attn = ATTN + (size_t)bh * HDIM * HDIM;

  v8f acc = {};
#pragma unroll
  for (int k = 0; k < HDIM; k += 4) {
    const int kk = k + half * 2;
    v2f a = *(const v2f*)(qrow + kk);
    v2f bfr;
    bfr.x = attn[(kk    ) * HDIM + n0 + lid];
    bfr.y = attn[(kk + 1) * HDIM + n0 + lid];
    acc = __builtin_amdgcn_wmma_f32_16x16x4_f32(false, a, false, bfr,
                                                (short)0, acc, false, false);
  }

#pragma unroll
  for (int r = 0; r < 8; ++r) {
    const int mr = m0 + r + half * 8;
    CTX[((size_t)b * NSEQ + mr) * DIM + h * HDIM + n0 + lid] = acc[r];
  }
}

// ---------------------------------------------------------------------------
extern "C" void kernel_launch(void* const* d_in, const int* in_sizes, int n_in,
                              void* d_out, int out_size, void* d_ws, size_t ws_size,
                              hipStream_t stream) {
  (void)in_sizes; (void)n_in; (void)out_size; (void)ws_size;
  const float* x     = (const float*)d_in[0];
  const float* wq    = (const float*)d_in[1];
  const float* wk    = (const float*)d_in[2];
  const float* wv    = (const float*)d_in[3];
  const float* wo    = (const float*)d_in[4];
  const float* bo    = (const float*)d_in[5];
  const float* alpha = (const float*)d_in[6];

  float* ws = (float*)d_ws;
  const size_t MN = (size_t)MTOT * DIM;
  float* q    = ws;
  float* k    = ws + MN;
  float* v    = ws + 2 * MN;
  float* ctx  = ws + 3 * MN;
  float* kkb  = ws + 4 * MN;
  float* ktvb = kkb  + (size_t)BATCH * HEADS * HDIM * HDIM;
  float* attn = ktvb + (size_t)BATCH * HEADS * HDIM * HDIM;

  // Fused QKV: per-wave 16x32x3 tiles; grid (M/128, N/32)
  qkv_gemm<<<dim3(MTOT / 128, DIM / 32), 256, 0, stream>>>(x, wq, wk, wv, q, k, v);

  gram_kernel<<<BATCH * HEADS, 512, 0, stream>>>(k, v, kkb, ktvb);

  inv_soft_kernel<<<BATCH * HEADS, 64, 0, stream>>>(kkb, ktvb, alpha, attn);

  apply_attn<<<dim3(NSEQ / 16, BATCH * HEADS), 128, 0, stream>>>(q, attn, ctx);

  out_gemm<<<dim3(MTOT / 128, DIM / 64), 256, 0, stream>>>(ctx, wo, bo, (float*)d_out);
}